// ParallelVarPatchEmbed_91182155694519
// MI455X (gfx1250) — compile-verified
//
#include <hip/hip_runtime.h>
#include <hip/hip_bf16.h>

// ParallelVarPatchEmbed: patchify + per-variable GEMM, fp32 WMMA 16x16x4.
//
// Shapes (hardcoded from reference):
//   x            [B=4, V=16, H=128, W=256]    fp32
//   vars         [16]                          int32 (indices into 48)
//   proj_weights [48, E=1024, 1, P=4, P=4]     fp32
//   proj_biases  [48, E=1024]                  fp32
//   out          [B, V, 2048, 1024]            fp32
//
// Per (b,v): out[patch, e] = sum_k A[patch,k] * Wv[k,e] + bias_v[e]
//   patch grid 32x64 (gh x gw), K = p*4+q (16), E = 1024.
//
// Wave tiling: one wave computes a 16(patch) x 16(embed) tile via 4 chained
// V_WMMA_F32_16X16X4_F32 (K = 16). Accumulator seeded with bias.
//
// WMMA f32 16x16x4 VGPR layouts (ISA 7.12.2):
//   A 16x4:  lanes 0-15 -> M, vgpr0/1 = K{0,1}; lanes 16-31 -> K{2,3}
//            => lane L: m = L%16, a = {A[m][kc+2*(L/16)], A[m][kc+2*(L/16)+1]}
//   B 4x16:  lane L: n = L%16, b = {B[kc+2*(L/16)][n], B[kc+2*(L/16)+1][n]}
//   C/D:     lane L: n = L%16, vgpr r -> m = r + 8*(L/16)

typedef __attribute__((ext_vector_type(2))) float v2f;
typedef __attribute__((ext_vector_type(8))) float v8f;

#define B_N   4
#define V_N   16
#define H_N   128
#define W_N   256
#define P_N   4
#define E_N   1024
#define GW_N  64            // W/P
#define NP_N  2048          // (H/P)*(W/P)
#define KTOT  16            // P*P

__global__ __launch_bounds__(256) void
patch_embed_wmma_f32(const float* __restrict__ x,
                     const int*   __restrict__ vars,
                     const float* __restrict__ wgt,   // [48][E][16]
                     const float* __restrict__ bias,  // [48][E]
                     float* __restrict__ out)         // [B*V][2048][E]
{
    const int lane  = threadIdx.x & 31;
    const int wave  = threadIdx.x >> 5;          // 0..7
    const int ptile = blockIdx.x;                // 0..127 (2048/16 patch tiles)
    const int bv    = blockIdx.y;                // 0..63  (B*V)
    const int v     = bv & (V_N - 1);

    const int var = vars[v];

    const int n = lane & 15;                     // N index / A's M index
    const int h = lane >> 4;                     // half-wave selector

    // ---- Load A tile: 16 patches x K=16, per-lane float2 per K-chunk ----
    const int   patch = ptile * 16 + n;          // this lane's patch row (M)
    const int   hg    = patch >> 6;              // patch / GW
    const int   wg    = patch & (GW_N - 1);
    const float* xb   = x + (size_t)bv * (H_N * W_N);

    v2f a[4];
#pragma unroll
    for (int kc = 0; kc < 4; ++kc) {
        const int k = kc * 4 + 2 * h;            // k%4 in {0,2} -> q,q+1 contiguous
        const int p = k >> 2;
        const int q = k & 3;
        const float* src = xb + (size_t)(hg * P_N + p) * W_N + (wg * P_N + q);
        a[kc] = *(const v2f*)src;                // 8B aligned (even element offset)
    }

    const float* wb = wgt  + (size_t)var * E_N * KTOT;
    const float* bb = bias + (size_t)var * E_N;
    float*       ob = out  + ((size_t)bv * NP_N + (size_t)ptile * 16) * E_N;

    // ---- 8 e-tiles per wave: wave w covers e-tiles w*8 .. w*8+7 ----
#pragma unroll 1
    for (int i = 0; i < 8; ++i) {
        const int e0 = (wave * 8 + i) * 16;

        // Seed accumulator with bias (depends only on n)
        const float bval = bb[e0 + n];
        v8f c;
#pragma unroll
        for (int r = 0; r < 8; ++r) c[r] = bval;

        // 4 chained fp32 WMMAs: K = 16
#pragma unroll
        for (int kc = 0; kc < 4; ++kc) {
            const int k = kc * 4 + 2 * h;
            v2f bm = *(const v2f*)(wb + (size_t)(e0 + n) * KTOT + k);  // 8B aligned
            c = __builtin_amdgcn_wmma_f32_16x16x4_f32(
                    /*neg_a=*/false, a[kc],
                    /*neg_b=*/false, bm,
                    /*c_mod=*/(short)0, c,
                    /*reuse_a=*/false, /*reuse_b=*/false);
        }

        // Store: vgpr r -> patch row (r + 8*h), embed e0+n.
        // Each half-wave writes 64B contiguous per row.
#pragma unroll
        for (int r = 0; r < 8; ++r) {
            const int pm = r + 8 * h;
            ob[(size_t)pm * E_N + e0 + n] = c[r];
        }
    }
}

extern "C" void kernel_launch(void* const* d_in, const int* in_sizes, int n_in,
                              void* d_out, int out_size, void* d_ws, size_t ws_size,
                              hipStream_t stream) {
    const float* x    = (const float*)d_in[0];
    const int*   vars = (const int*)d_in[1];
    const float* wgt  = (const float*)d_in[2];
    const float* bias = (const float*)d_in[3];
    float*       out  = (float*)d_out;

    dim3 grid(NP_N / 16, B_N * V_N);   // 128 x 64
    dim3 block(256);                   // 8 waves
    patch_embed_wmma_f32<<<grid, block, 0, stream>>>(x, vars, wgt, bias, out);
}